// GAT_28037546508640
// MI455X (gfx1250) — compile-verified
//
#include <hip/hip_runtime.h>
#include <math.h>

#define NN  50000
#define EE  800000
#define DIN 128
#define HH  2
#define FF  64
#define DHF 128   /* H*F == DIN */
#define NEG_SLOPE 0.2f

typedef __attribute__((ext_vector_type(2))) float v2f;
typedef __attribute__((ext_vector_type(8))) float v8f;

#if defined(__AMDGCN__) && __has_builtin(__builtin_amdgcn_wmma_f32_16x16x4_f32)
#define USE_WMMA_F32 1
#else
#define USE_WMMA_F32 0
#endif

// ---------------------------------------------------------------------------
// feat[NN,128] = h[NN,128] @ W[128,128]  via V_WMMA_F32_16X16X4_F32
// Block = 256 threads = 8 waves; wave w computes rows [16*bx,16*bx+16) x cols [16w,16w+16)
// A tile staged in LDS (row stride 132 to break bank conflicts), B streamed from L2.
// ---------------------------------------------------------------------------
__global__ __launch_bounds__(256) void gat_gemm(const float* __restrict__ Hm,
                                                const float* __restrict__ W,
                                                float* __restrict__ feat)
{
    __shared__ float sA[16 * 132];
    const int tid  = threadIdx.x;
    const int row0 = blockIdx.x * 16;

    for (int i = tid; i < 16 * DIN; i += 256) {
        int r = i >> 7, c = i & 127;
        int gr = row0 + r;
        sA[r * 132 + c] = (gr < NN) ? Hm[gr * DIN + c] : 0.0f;
    }
    __syncthreads();

    const int wave = tid >> 5;
    const int lane = tid & 31;
    const int mrow = lane & 15;
    const int hi   = lane >> 4;      // 0: lanes 0-15 (K+0,K+1), 1: lanes 16-31 (K+2,K+3)
    const int col0 = wave * 16;

    v8f c = {};
#if USE_WMMA_F32
    for (int k = 0; k < DIN; k += 4) {
        const int ka = k + 2 * hi;
        v2f a, b;
        a.x = sA[mrow * 132 + ka + 0];
        a.y = sA[mrow * 132 + ka + 1];
        b.x = W[(ka + 0) * DHF + col0 + mrow];
        b.y = W[(ka + 1) * DHF + col0 + mrow];
        c = __builtin_amdgcn_wmma_f32_16x16x4_f32(false, a, false, b,
                                                  (short)0, c, false, false);
    }
#else
    for (int k = 0; k < DIN; ++k) {
        float bkn = W[k * DHF + col0 + mrow];
        #pragma unroll
        for (int i = 0; i < 8; ++i)
            c[i] = fmaf(sA[(i + 8 * hi) * 132 + k], bkn, c[i]);
    }
#endif
    #pragma unroll
    for (int i = 0; i < 8; ++i) {
        int M = i + 8 * hi;                 // C/D layout: VGPR i -> row i (+8 for high half)
        int r = row0 + M;
        if (r < NN) feat[r * DHF + col0 + mrow] = c[i];
    }
}

// el/er node attention scores: one thread per (node, head), 64-wide dot
__global__ __launch_bounds__(256) void gat_attn(const float* __restrict__ feat,
                                                const float* __restrict__ al,
                                                const float* __restrict__ ar,
                                                float* __restrict__ el,
                                                float* __restrict__ er)
{
    int i = blockIdx.x * 256 + threadIdx.x;
    if (i >= NN * HH) return;
    int n = i >> 1, h = i & 1;
    const float* fp  = feat + n * DHF + h * FF;
    const float* alp = al + h * FF;
    const float* arp = ar + h * FF;
    float sl = 0.f, sr = 0.f;
    #pragma unroll 8
    for (int f = 0; f < FF; ++f) {
        float v = fp[f];
        sl = fmaf(v, alp[f], sl);
        sr = fmaf(v, arp[f], sr);
    }
    el[i] = sl; er[i] = sr;
}

// ------------------------- CSR build (runs once per call) -------------------
__global__ __launch_bounds__(256) void csr_zero(int* __restrict__ deg)
{
    int i = blockIdx.x * 256 + threadIdx.x;
    if (i < NN) deg[i] = 0;
}

__global__ __launch_bounds__(256) void csr_hist(const int* __restrict__ dst,
                                                int* __restrict__ deg)
{
    int e = blockIdx.x * 256 + threadIdx.x;
    if (e < EE) atomicAdd(&deg[dst[e]], 1);
}

// single-block exclusive scan over deg[NN] -> off[NN+1]; cursor = copy of off
__global__ __launch_bounds__(1024) void csr_scan(const int* __restrict__ deg,
                                                 int* __restrict__ off,
                                                 int* __restrict__ cursor)
{
    __shared__ int buf[1024];
    __shared__ int carry;
    const int t = threadIdx.x;
    if (t == 0) carry = 0;
    __syncthreads();
    for (int base = 0; base < NN; base += 1024) {
        int i = base + t;
        int v = (i < NN) ? deg[i] : 0;
        buf[t] = v;
        __syncthreads();
        for (int ofs = 1; ofs < 1024; ofs <<= 1) {
            int x = (t >= ofs) ? buf[t - ofs] : 0;
            __syncthreads();
            buf[t] += x;
            __syncthreads();
        }
        int excl = buf[t] - v + carry;
        if (i < NN) { off[i] = excl; cursor[i] = excl; }
        __syncthreads();
        if (t == 1023) carry += buf[1023];
        __syncthreads();
    }
    if (t == 0) off[NN] = carry;   // == EE
}

__global__ __launch_bounds__(256) void csr_build(const int* __restrict__ src,
                                                 const int* __restrict__ dst,
                                                 int* __restrict__ cursor,
                                                 int* __restrict__ csrc,
                                                 int* __restrict__ cdst)
{
    int e = blockIdx.x * 256 + threadIdx.x;
    if (e >= EE) return;
    int d = dst[e];
    int pos = atomicAdd(&cursor[d], 1);
    csrc[pos] = src[e];
    cdst[pos] = d;
}

// ------------------- per-layer edge scores in CSR order ---------------------
__global__ __launch_bounds__(256) void gat_edge_e(const float* __restrict__ el,
                                                  const float* __restrict__ er,
                                                  const int* __restrict__ csrc,
                                                  const int* __restrict__ cdst,
                                                  float* __restrict__ ecsr)
{
    int i = blockIdx.x * 256 + threadIdx.x;
    if (i >= EE * HH) return;
    int pos = i >> 1, h = i & 1;
    int s = csrc[pos], d = cdst[pos];
    float v = el[s * HH + h] + er[d * HH + h];
    ecsr[i] = (v >= 0.f) ? v : NEG_SLOPE * v;
}

// ---------------- atomic-free segment softmax + aggregation -----------------
// One block (128 thr = 4 waves) per destination node. Thread t owns feature t
// (head = t/64). Pass 1: max(e). Pass 2: den = sum exp(e-m) and
// acc = sum exp(e-m)*feat[src][t]; then rst = acc/den (alpha folds out),
// fused bias + ELU (+ head mean on the last layer).
__global__ __launch_bounds__(128) void gat_aggregate(const float* __restrict__ ecsr,
                                                     const float* __restrict__ feat,
                                                     const int* __restrict__ csrc,
                                                     const int* __restrict__ off,
                                                     const float* __restrict__ bias,
                                                     float* __restrict__ out,
                                                     int last)
{
    __shared__ float sv[DHF];
    const int d = blockIdx.x;
    const int t = threadIdx.x;
    const int h = t >> 6;
    const int p0 = off[d], p1 = off[d + 1];

    float m = -__builtin_inff();
    for (int p = p0; p < p1; ++p)
        m = fmaxf(m, ecsr[p * HH + h]);

    float den = 0.f, acc = 0.f;
    for (int p = p0; p < p1; ++p) {
        float ex = expf(ecsr[p * HH + h] - m);
        den += ex;
        int s = csrc[p];
        acc = fmaf(ex, feat[s * DHF + t], acc);
    }
    float r = acc / ((den == 0.f) ? 1.f : den);
    float v = r + bias[t];
    v = (v > 0.f) ? v : expm1f(v);

    if (!last) {
        out[d * DHF + t] = v;
    } else {
        sv[t] = v;
        __syncthreads();
        if (t < FF) out[d * FF + t] = 0.5f * (sv[t] + sv[t + FF]);
    }
}

extern "C" void kernel_launch(void* const* d_in, const int* in_sizes, int n_in,
                              void* d_out, int out_size, void* d_ws, size_t ws_size,
                              hipStream_t stream)
{
    const float* x    = (const float*)d_in[0];
    const int*   src  = (const int*)d_in[1];
    const int*   dst  = (const int*)d_in[2];
    const float* Wall = (const float*)d_in[3];   // [3,128,128]
    const float* al   = (const float*)d_in[4];   // [3,2,64]
    const float* ar   = (const float*)d_in[5];   // [3,2,64]
    const float* bias = (const float*)d_in[6];   // [3,2,64]
    float* out = (float*)d_out;

    char* p = (char*)d_ws;
    auto take = [&](size_t bytes) {
        char* q = p;
        p += (bytes + 255) & ~size_t(255);
        return q;
    };
    float* feat   = (float*)take(sizeof(float) * NN * DHF);
    float* hbuf   = (float*)take(sizeof(float) * NN * DHF);
    float* el     = (float*)take(sizeof(float) * NN * HH);
    float* er     = (float*)take(sizeof(float) * NN * HH);
    float* ecsr   = (float*)take(sizeof(float) * EE * HH);
    int*   deg    = (int*)take(sizeof(int) * NN);
    int*   off    = (int*)take(sizeof(int) * (NN + 1));
    int*   cursor = (int*)take(sizeof(int) * NN);
    int*   csrc   = (int*)take(sizeof(int) * EE);
    int*   cdst   = (int*)take(sizeof(int) * EE);

    const int tb   = 256;
    const int gN   = (NN + tb - 1) / tb;
    const int gE   = (EE + tb - 1) / tb;
    const int gN2  = (NN * HH + tb - 1) / tb;
    const int gE2  = (EE * HH + tb - 1) / tb;
    const int gGmm = (NN + 15) / 16;

    // ---- CSR of incoming edges: built once, reused by all 3 layers ----
    csr_zero <<<gN, tb, 0, stream>>>(deg);
    csr_hist <<<gE, tb, 0, stream>>>(dst, deg);
    csr_scan <<<1, 1024, 0, stream>>>(deg, off, cursor);
    csr_build<<<gE, tb, 0, stream>>>(src, dst, cursor, csrc, cdst);

    const float* hin = x;
    for (int layer = 0; layer < 3; ++layer) {
        const float* Wl  = Wall + layer * DIN * DHF;
        const float* all = al   + layer * HH * FF;
        const float* arl = ar   + layer * HH * FF;
        const float* bl  = bias + layer * HH * FF;
        const int    last = (layer == 2);
        float* outp = last ? out : hbuf;

        gat_gemm     <<<gGmm, tb, 0, stream>>>(hin, Wl, feat);
        gat_attn     <<<gN2,  tb, 0, stream>>>(feat, all, arl, el, er);
        gat_edge_e   <<<gE2,  tb, 0, stream>>>(el, er, csrc, cdst, ecsr);
        gat_aggregate<<<NN,  128, 0, stream>>>(ecsr, feat, csrc, off, bl, outp, last);
        hin = hbuf;
    }
    (void)in_sizes; (void)n_in; (void)out_size; (void)ws_size;
}